// Net_31894427140749
// MI455X (gfx1250) — compile-verified
//
#include <hip/hip_runtime.h>

typedef float v2f __attribute__((ext_vector_type(2)));
typedef float v8f __attribute__((ext_vector_type(8)));

#define TPB 256

__global__ void k_fill(float* __restrict__ p, float v, int n) {
  int i = blockIdx.x * blockDim.x + threadIdx.x;
  if (i < n) p[i] = v;
}

__global__ void k_deg_accum(const int* __restrict__ dst, float* __restrict__ deg, int E) {
  int e = blockIdx.x * blockDim.x + threadIdx.x;
  if (e < E) atomicAdd(&deg[dst[e]], 1.0f);
}

__global__ void k_rsqrt_inplace(float* __restrict__ d, int n) {
  int i = blockIdx.x * blockDim.x + threadIdx.x;
  if (i < n) d[i] = rsqrtf(d[i]);   // deg >= 1 always (self-loop)
}

__global__ void k_edge_norm(const int* __restrict__ src, const int* __restrict__ dst,
                            const float* __restrict__ dinv, float* __restrict__ norm, int E) {
  int e = blockIdx.x * blockDim.x + threadIdx.x;
  if (e < E) norm[e] = dinv[src[e]] * dinv[dst[e]];
}

// P[N,16] = x[N,4] @ W[4,16] — one V_WMMA_F32_16X16X4_F32 per 16-node tile per wave.
__global__ void k_gemm4_wmma(const float* __restrict__ x, const float* __restrict__ W,
                             float* __restrict__ out, int ntiles) {
  int wave = (blockIdx.x * blockDim.x + threadIdx.x) >> 5;
  int lane = threadIdx.x & 31;
  if (wave >= ntiles) return;               // wave-uniform guard: EXEC all-1s inside
  int base = wave * 16;
  int m    = lane & 15;                     // row (A) / col (B,D)
  int kb   = (lane >> 4) * 2;               // lanes 0-15: K=0,1 ; lanes 16-31: K=2,3
  const float* xr = x + (size_t)(base + m) * 4;
  v2f a; a.x = xr[kb];             a.y = xr[kb + 1];
  v2f b; b.x = W[kb * 16 + m];     b.y = W[(kb + 1) * 16 + m];
  v8f c = {};
  c = __builtin_amdgcn_wmma_f32_16x16x4_f32(false, a, false, b, (short)0, c, false, false);
  int rowoff = (lane >> 4) * 8;             // D: lanes 16-31 hold M = v+8
  float* o = out + (size_t)base * 16 + m;
#pragma unroll
  for (int v = 0; v < 8; ++v) o[(size_t)(v + rowoff) * 16] = c[v];
}

// P[N,16] = relu(AGG[N,16] + bias) @ W[16,16] — bias+relu fused into the A-fragment
// load (each AGG element is read exactly once per tile), 4 chained K=4 fp32 WMMAs.
__global__ void k_gemm16_fused_wmma(const float* __restrict__ agg, const float* __restrict__ bias,
                                    const float* __restrict__ W, float* __restrict__ out,
                                    int ntiles) {
  int wave = (blockIdx.x * blockDim.x + threadIdx.x) >> 5;
  int lane = threadIdx.x & 31;
  if (wave >= ntiles) return;
  int base = wave * 16;
  int m    = lane & 15;
  int khb  = (lane >> 4) * 2;
  const float* hr = agg + (size_t)(base + m) * 16;
  v8f c = {};
#pragma unroll
  for (int kk = 0; kk < 4; ++kk) {
    int k0 = kk * 4 + khb;
    v2f a; a.x = fmaxf(hr[k0]     + bias[k0],     0.f);
           a.y = fmaxf(hr[k0 + 1] + bias[k0 + 1], 0.f);
    v2f b; b.x = W[k0 * 16 + m];   b.y = W[(k0 + 1) * 16 + m];
    c = __builtin_amdgcn_wmma_f32_16x16x4_f32(false, a, false, b, (short)0, c, false, false);
  }
  int rowoff = (lane >> 4) * 8;
  float* o = out + (size_t)base * 16 + m;
#pragma unroll
  for (int v = 0; v < 8; ++v) o[(size_t)(v + rowoff) * 16] = c[v];
}

// Scalar tails in case N % 16 != 0 (N=100000 is a multiple of 16, so these normally don't launch).
__global__ void k_gemm4_tail(const float* __restrict__ x, const float* __restrict__ W,
                             float* __restrict__ out, int start, int N) {
  int t = blockIdx.x * blockDim.x + threadIdx.x;
  int node = start + (t >> 4), c = t & 15;
  if (node < N) {
    float s = 0.f;
    for (int k = 0; k < 4; ++k) s += x[node * 4 + k] * W[k * 16 + c];
    out[node * 16 + c] = s;
  }
}
__global__ void k_gemm16_fused_tail(const float* __restrict__ agg, const float* __restrict__ bias,
                                    const float* __restrict__ W, float* __restrict__ out,
                                    int start, int N) {
  int t = blockIdx.x * blockDim.x + threadIdx.x;
  int node = start + (t >> 4), c = t & 15;
  if (node < N) {
    float s = 0.f;
    for (int k = 0; k < 16; ++k)
      s += fmaxf(agg[node * 16 + k] + bias[k], 0.f) * W[k * 16 + c];
    out[node * 16 + c] = s;
  }
}

// AGG init = P * dinv^2 (self-loop contribution; zero-free init, costs no atomics)
__global__ void k_self_init(const float* __restrict__ P, const float* __restrict__ dinv,
                            float* __restrict__ agg, int N) {
  int t = blockIdx.x * blockDim.x + threadIdx.x;
  if (t < N * 16) {
    float d = dinv[t >> 4];
    agg[t] = P[t] * d * d;
  }
}

// AGG[dst] += P[src] * norm, thread = (edge, channel): coalesced 64B groups per edge.
__global__ void k_scatter(const int* __restrict__ src, const int* __restrict__ dst,
                          const float* __restrict__ norm, const float* __restrict__ P,
                          float* __restrict__ agg, int E) {
  int t = blockIdx.x * blockDim.x + threadIdx.x;
  if (t < E * 16) {
    int e = t >> 4, c = t & 15;
    float v = P[(size_t)src[e] * 16 + c] * norm[e];
    atomicAdd(&agg[(size_t)dst[e] * 16 + c], v);
  }
}

// global_add_pool fused with the last layer's bias+relu: g[batch[i]] += relu(AGG3 + b3)
__global__ void k_pool_biasrelu(const float* __restrict__ agg, const float* __restrict__ bias,
                                const int* __restrict__ batch, float* __restrict__ g, int N) {
  int t = blockIdx.x * blockDim.x + threadIdx.x;
  if (t < N * 16) {
    int i = t >> 4, c = t & 15;
    atomicAdd(&g[(size_t)batch[i] * 16 + c], fmaxf(agg[t] + bias[c], 0.f));
  }
}

__global__ void k_mlp(const float* __restrict__ g, const float* __restrict__ y,
                      const float* __restrict__ fcW1, const float* __restrict__ fcb1,
                      const float* __restrict__ fcW2, const float* __restrict__ fcb2,
                      const float* __restrict__ fcW3, const float* __restrict__ fcb3,
                      float* __restrict__ out, int G) {
  int r = blockIdx.x * blockDim.x + threadIdx.x;
  if (r >= G) return;
  float feat[20];
#pragma unroll
  for (int k = 0; k < 16; ++k) feat[k] = g[r * 16 + k];
#pragma unroll
  for (int k = 0; k < 4; ++k) feat[16 + k] = y[r * 4 + k];
  float t1[16];
#pragma unroll
  for (int j = 0; j < 16; ++j) {
    float s = fcb1[j];
    for (int k = 0; k < 20; ++k) s += feat[k] * fcW1[k * 16 + j];
    t1[j] = fmaxf(s, 0.f);
  }
  float t2[16];
#pragma unroll
  for (int j = 0; j < 16; ++j) {
    float s = fcb2[j];
    for (int k = 0; k < 16; ++k) s += t1[k] * fcW2[k * 16 + j];
    t2[j] = fmaxf(s, 0.f);
  }
  float s = fcb3[0];
#pragma unroll
  for (int j = 0; j < 16; ++j) s += t2[j] * fcW3[j];
  out[r] = s;
}

static inline int grid_for(long long n, int b) { return (int)((n + b - 1) / b); }

extern "C" void kernel_launch(void* const* d_in, const int* in_sizes, int n_in,
                              void* d_out, int out_size, void* d_ws, size_t ws_size,
                              hipStream_t stream) {
  const float* x     = (const float*)d_in[0];
  const int*   ei    = (const int*)  d_in[1];   // [2, E] int32 (JAX x64 disabled)
  const float* y     = (const float*)d_in[2];
  const int*   batch = (const int*)  d_in[3];
  const float* W1 = (const float*)d_in[4],  *b1 = (const float*)d_in[5];
  const float* W2 = (const float*)d_in[6],  *b2 = (const float*)d_in[7];
  const float* W3 = (const float*)d_in[8],  *b3 = (const float*)d_in[9];
  const float* fcW1 = (const float*)d_in[10], *fcb1 = (const float*)d_in[11];
  const float* fcW2 = (const float*)d_in[12], *fcb2 = (const float*)d_in[13];
  const float* fcW3 = (const float*)d_in[14], *fcb3 = (const float*)d_in[15];
  float* out = (float*)d_out;

  const int N = in_sizes[0] / 4;
  const int E = in_sizes[1] / 2;
  const int G = in_sizes[2] / 4;
  const int* src = ei;
  const int* dst = ei + E;

  float* ws   = (float*)d_ws;
  float* dinv = ws;  ws += N;
  float* norm = ws;  ws += E;
  float* hbuf = ws;  ws += (size_t)N * 16;   // P_l  (gemm output)
  float* abuf = ws;  ws += (size_t)N * 16;   // AGG_l
  float* g    = ws;  ws += (size_t)G * 16;

  const int ntiles = N / 16;
  const int rem    = N % 16;

  // --- GCN normalization: deg (init 1.0 = self-loop) -> dinv -> per-edge norm (reused 3x) ---
  k_fill<<<grid_for(N, TPB), TPB, 0, stream>>>(dinv, 1.0f, N);
  k_deg_accum<<<grid_for(E, TPB), TPB, 0, stream>>>(dst, dinv, E);
  k_rsqrt_inplace<<<grid_for(N, TPB), TPB, 0, stream>>>(dinv, N);
  k_edge_norm<<<grid_for(E, TPB), TPB, 0, stream>>>(src, dst, dinv, norm, E);

  const int gw = grid_for(ntiles, TPB / 32);            // 8 waves per 256-thread block
  const int gE = grid_for((long long)E * 16, TPB);
  const int gN = grid_for((long long)N * 16, TPB);

  // --- Layer 1: P1 = x@W1 ; AGG1 = selfloop + edge scatter ---
  k_gemm4_wmma<<<gw, TPB, 0, stream>>>(x, W1, hbuf, ntiles);
  if (rem) k_gemm4_tail<<<grid_for((long long)rem * 16, TPB), TPB, 0, stream>>>(x, W1, hbuf, ntiles * 16, N);
  k_self_init<<<gN, TPB, 0, stream>>>(hbuf, dinv, abuf, N);
  k_scatter<<<gE, TPB, 0, stream>>>(src, dst, norm, hbuf, abuf, E);

  // --- Layer 2: P2 = relu(AGG1+b1)@W2 (fused) ; AGG2 ---
  k_gemm16_fused_wmma<<<gw, TPB, 0, stream>>>(abuf, b1, W2, hbuf, ntiles);
  if (rem) k_gemm16_fused_tail<<<grid_for((long long)rem * 16, TPB), TPB, 0, stream>>>(abuf, b1, W2, hbuf, ntiles * 16, N);
  k_self_init<<<gN, TPB, 0, stream>>>(hbuf, dinv, abuf, N);
  k_scatter<<<gE, TPB, 0, stream>>>(src, dst, norm, hbuf, abuf, E);

  // --- Layer 3: P3 = relu(AGG2+b2)@W3 (fused) ; AGG3 ---
  k_gemm16_fused_wmma<<<gw, TPB, 0, stream>>>(abuf, b2, W3, hbuf, ntiles);
  if (rem) k_gemm16_fused_tail<<<grid_for((long long)rem * 16, TPB), TPB, 0, stream>>>(abuf, b2, W3, hbuf, ntiles * 16, N);
  k_self_init<<<gN, TPB, 0, stream>>>(hbuf, dinv, abuf, N);
  k_scatter<<<gE, TPB, 0, stream>>>(src, dst, norm, hbuf, abuf, E);

  // --- pool (fused bias+relu of layer 3) + MLP head ---
  k_fill<<<grid_for(G * 16, TPB), TPB, 0, stream>>>(g, 0.0f, G * 16);
  k_pool_biasrelu<<<gN, TPB, 0, stream>>>(abuf, b3, batch, g, N);
  k_mlp<<<grid_for(G, TPB), TPB, 0, stream>>>(g, y, fcW1, fcb1, fcW2, fcb2, fcW3, fcb3, out, G);
}